// EdgeCostFunction_5179730559665
// MI455X (gfx1250) — compile-verified
//
#include <hip/hip_runtime.h>
#include <hip/hip_bf16.h>
#include <math.h>

// Problem constants (from reference)
#define B_ 16
#define Q_ 200
#define N_ 200
#define C_ 64
#define T_ 100
#define G_ 50
#define S_ 4
#define E_ (G_ * S_)      // 200 edges per batch
#define TILES_ 7          // ceil(100/16) row tiles of 16

typedef __attribute__((ext_vector_type(2))) float v2f;
typedef __attribute__((ext_vector_type(8))) float v8f;

#if defined(__has_builtin)
#if __has_builtin(__builtin_amdgcn_wmma_f32_16x16x4_f32)
#define USE_WMMA_F32 1
#endif
#endif

// ---------------------------------------------------------------------------
// Per-batch prep: stable argsort of target_indices (T=100, O(T^2) rank),
// sq = query_indices[order], and resolve per-edge gather indices:
//   t_idx[e] = col_ids[e]           (denominator slot)
//   off[e]   = sq[t]*C + class      (element offset within the (b,q) slab)
// ---------------------------------------------------------------------------
__global__ __launch_bounds__(128) void prep_kernel(
    const int* __restrict__ query_indices,   // [B,T]
    const int* __restrict__ target_indices,  // [B,T]
    const int* __restrict__ col_ids,         // [B,E]
    const int* __restrict__ edge_tgt,        // [B,E]
    int* __restrict__ ws_sq,                 // [B,T]
    int* __restrict__ ws_t,                  // [B,E]
    int* __restrict__ ws_off)                // [B,E]
{
    __shared__ int s_key[T_];
    __shared__ int s_qi[T_];
    __shared__ int s_sq[T_];
    const int b = blockIdx.x;
    const int tid = threadIdx.x;

    if (tid < T_) {
        s_key[tid] = target_indices[b * T_ + tid];
        s_qi[tid]  = query_indices[b * T_ + tid];
    }
    __syncthreads();
    if (tid < T_) {
        const int k = s_key[tid];
        int rank = 0;
        for (int j = 0; j < T_; ++j) {
            const int kj = s_key[j];
            rank += (kj < k) || (kj == k && j < tid);   // stable tie-break
        }
        s_sq[rank] = s_qi[tid];                          // sq[argsort pos]
    }
    __syncthreads();
    if (tid < T_) ws_sq[b * T_ + tid] = s_sq[tid];
    for (int e = tid; e < E_; e += blockDim.x) {
        const int t = col_ids[b * E_ + e];
        const int n = s_sq[t];
        const int c = edge_tgt[b * E_ + e];
        ws_t[b * E_ + e]   = t;
        ws_off[b * E_ + e] = n * C_ + c;
    }
}

// ---------------------------------------------------------------------------
// One block per (b,q). Phase A: 7 waves each compute 16 softmax row sums via
// a chain of 16 V_WMMA_F32_16X16X4_F32 with all-ones B (D = row-sum of exp).
// Wave 7 prefetches phase-B gather addresses. Phase B: 200 threads gather
// numerators; Phase C: group sums; Phase D: broadcast to all B output copies.
// ---------------------------------------------------------------------------
__global__ __launch_bounds__(256) void cost_kernel(
    const float* __restrict__ edges,   // [B,Q,N,C]
    const int* __restrict__ ws_sq,     // [B,T]
    const int* __restrict__ ws_t,      // [B,E]
    const int* __restrict__ ws_off,    // [B,E]
    float* __restrict__ out)           // [B,Q,B*G]
{
    __shared__ float s_rowmax[TILES_ * 16];   // 112
    __shared__ float s_denom[TILES_ * 16];    // 112
    __shared__ float s_pick[E_];
    __shared__ float s_cost[G_];

    const int q    = blockIdx.x;
    const int b    = blockIdx.y;
    const int tid  = threadIdx.x;
    const int wave = tid >> 5;
    const int lane = tid & 31;

    const float* __restrict__ bq = edges + (size_t)(b * Q_ + q) * (N_ * C_);

    if (wave < TILES_) {
        // A-matrix f32 16x4 layout: lanes 0-15 hold K={0,1}, lanes 16-31 K={2,3}
        const int half = lane >> 4;       // 0 or 1
        const int m    = lane & 15;       // row within tile
        const int r    = wave * 16 + m;   // global sorted-target row id

        v2f vals[16];
        if (r < T_) {
            const int n = ws_sq[b * T_ + r];
            const float* __restrict__ rowf = bq + n * C_;
#pragma unroll
            for (int k = 0; k < 16; ++k)
                vals[k] = *(const v2f*)(rowf + 4 * k + 2 * half);  // 8B aligned
        } else {
#pragma unroll
            for (int k = 0; k < 16; ++k) { vals[k].x = 0.f; vals[k].y = 0.f; }
        }

        // row max across this lane's 32 values + the partner half-lane
        float mx = vals[0].x;
#pragma unroll
        for (int k = 0; k < 16; ++k) {
            mx = fmaxf(mx, vals[k].x);
            mx = fmaxf(mx, vals[k].y);
        }
        mx = fmaxf(mx, __shfl_xor(mx, 16, 32));

#pragma unroll
        for (int k = 0; k < 16; ++k) {
            vals[k].x = expf(vals[k].x - mx);
            vals[k].y = expf(vals[k].y - mx);
        }

#if USE_WMMA_F32
        // denominators = row sums via 16 chained f32 WMMAs, B = ones (4x16)
        v8f acc = {0.f, 0.f, 0.f, 0.f, 0.f, 0.f, 0.f, 0.f};
        v2f ones; ones.x = 1.f; ones.y = 1.f;
#pragma unroll
        for (int k = 0; k < 16; ++k) {
            acc = __builtin_amdgcn_wmma_f32_16x16x4_f32(
                false, vals[k], false, ones, (short)0, acc, false, false);
        }
        // D layout: VGPR j -> row j (lanes 0-15) / row j+8 (lanes 16-31)
        if (lane == 0) {
#pragma unroll
            for (int j = 0; j < 8; ++j) s_denom[wave * 16 + j] = acc[j];
        }
        if (lane == 16) {
#pragma unroll
            for (int j = 0; j < 8; ++j) s_denom[wave * 16 + 8 + j] = acc[j];
        }
#else
        // fallback: lane-pair shuffle reduction
        float s = 0.f;
#pragma unroll
        for (int k = 0; k < 16; ++k) s += vals[k].x + vals[k].y;
        s += __shfl_xor(s, 16, 32);
        if (half == 0) s_denom[r] = s;
#endif
        if (half == 0) s_rowmax[r] = mx;
    } else {
        // wave 7: warm caches for the phase-B scattered gathers
        for (int e = lane; e < E_; e += 32) {
            const int off = ws_off[b * E_ + e];
            __builtin_prefetch(bq + off, 0, 0);   // -> global_prefetch_b8
        }
    }
    __syncthreads();

    // Phase B: picked[e] = exp(x - rowmax[t]) / denom[t]
    if (tid < E_) {
        const int t   = ws_t[b * E_ + tid];
        const int off = ws_off[b * E_ + tid];
        const float x = bq[off];
        s_pick[tid] = expf(x - s_rowmax[t]) / s_denom[t];
    }
    __syncthreads();

    // Phase C: cost[g] = -sum of group of 4
    if (tid < G_) {
        s_cost[tid] = -(s_pick[4 * tid] + s_pick[4 * tid + 1] +
                        s_pick[4 * tid + 2] + s_pick[4 * tid + 3]);
    }
    __syncthreads();

    // Phase D: out[b2, q, b*G + g] = cost[g] for every b2 (broadcast copies)
    for (int i = tid; i < B_ * G_; i += 256) {
        const int b2 = i / G_;
        const int g  = i % G_;
        out[((size_t)b2 * Q_ + q) * (B_ * G_) + (size_t)b * G_ + g] = s_cost[g];
    }
}

extern "C" void kernel_launch(void* const* d_in, const int* in_sizes, int n_in,
                              void* d_out, int out_size, void* d_ws, size_t ws_size,
                              hipStream_t stream) {
    (void)in_sizes; (void)n_in; (void)out_size; (void)ws_size;
    const float* edges = (const float*)d_in[0];   // batch_edges [B,Q,N,C] f32
    const int*   qidx  = (const int*)d_in[1];     // query_indices [B,T]
    const int*   tkey  = (const int*)d_in[2];     // target_indices [B,T]
    const int*   col   = (const int*)d_in[3];     // col_ids [B,G,S]
    const int*   etgt  = (const int*)d_in[4];     // edge_target_indices [B,G,S]

    int* ws_sq  = (int*)d_ws;            // B*T  = 1600 ints
    int* ws_t   = ws_sq + B_ * T_;       // B*E  = 3200 ints
    int* ws_off = ws_t + B_ * E_;        // B*E  = 3200 ints

    prep_kernel<<<B_, 128, 0, stream>>>(qidx, tkey, col, etgt, ws_sq, ws_t, ws_off);

    dim3 grid(Q_, B_);
    cost_kernel<<<grid, 256, 0, stream>>>(edges, ws_sq, ws_t, ws_off, (float*)d_out);
}